// HeteroGNN_68624987455883
// MI455X (gfx1250) — compile-verified
//
#include <hip/hip_runtime.h>
#include <hip/hip_bf16.h>
#include <math.h>

#define HDIM   256
#define OUTC   128
#define LAYERS 4
#define RELS   4
#define APAD   260   // LDS row stride in floats (kills bank conflicts: 260 % 64 = 4)

typedef __attribute__((ext_vector_type(2))) float v2f;
typedef __attribute__((ext_vector_type(8))) float v8f;

// ---------------- scatter-add: one wave (32 lanes) per edge ----------------
// agg[dst] += x_src[src]  (coalesced: lane i handles columns i, i+32, ...)
// deg[dst] += 1
__global__ void hg_scatter(const float* __restrict__ xsrc,
                           const int* __restrict__ src,
                           const int* __restrict__ dst,
                           float* __restrict__ agg,
                           float* __restrict__ deg,
                           int nEdges)
{
    int gwave = (int)((blockIdx.x * blockDim.x + threadIdx.x) >> 5);
    int lane  = threadIdx.x & 31;
    if (gwave >= nEdges) return;
    int s = src[gwave];
    int d = dst[gwave];
    const float* xs = xsrc + (size_t)s * HDIM;
    float* ag = agg + (size_t)d * HDIM;
#pragma unroll
    for (int c = lane; c < HDIM; c += 32)
        atomicAdd(&ag[c], xs[c]);
    if (lane == 0) atomicAdd(&deg[d], 1.0f);
}

// deg[i] = 1 / max(deg[i], 1)
__global__ void hg_invdeg(float* __restrict__ deg, int n)
{
    int i = blockIdx.x * blockDim.x + threadIdx.x;
    if (i < n) deg[i] = 1.0f / fmaxf(deg[i], 1.0f);
}

// ---------------- fp32 WMMA GEMM: C[rows x Hout] (+)= (A*rowScale) @ W^T + bias
// A: [rows x 256] row-major, W: [Hout x 256] row-major.
// blockDim.x = 32 * (Hout/64); each wave computes a 16x64 tile with 4
// v_wmma_f32_16x16x4_f32 accumulators, looping K=256 in steps of 4.
// The 16x256 A tile is staged once per block into LDS with
// global_load_async_to_lds_b128 (coalesced 16B/lane), then A fragments are
// fed from LDS (ds_load_b64, bank-conflict-free via padded row stride).
__global__ void hg_gemm_wmma(const float* __restrict__ A,
                             const float* __restrict__ W,
                             const float* __restrict__ bias,     // nullable
                             const float* __restrict__ rowScale, // nullable
                             float* __restrict__ C,
                             int Hout, int accumulate)
{
    __shared__ float As[16 * APAD];

    const int lane = threadIdx.x & 31;
    const int wave = threadIdx.x >> 5;
    const int l16  = lane & 15;
    const int hi   = lane >> 4;

    const int row0    = blockIdx.x * 16;
    const int colBase = wave * 64;

    // ---- async stage of the A tile: 16 rows x 256 floats = 1024 x 16B chunks
    {
        const int tid = threadIdx.x;
        const int nth = blockDim.x;
        for (int c = tid; c < 1024; c += nth) {
            const int r = c >> 6;              // row within tile (64 chunks/row)
            const int o = (c & 63) << 2;       // float offset within row
            const float* g = A + (size_t)(row0 + r) * HDIM + o;
            unsigned lds = (unsigned)(uintptr_t)(&As[r * APAD + o]);
            asm volatile("global_load_async_to_lds_b128 %0, %1, off"
                         :: "v"(lds), "v"(g) : "memory");
        }
        asm volatile("s_wait_asynccnt 0x0" ::: "memory");
        __syncthreads();
    }

    const float scale = rowScale ? rowScale[row0 + l16] : 1.0f;
    const float* __restrict__ arowp = &As[l16 * APAD];

    const float* __restrict__ w0 = W + (size_t)(colBase +  0 + l16) * HDIM;
    const float* __restrict__ w1 = W + (size_t)(colBase + 16 + l16) * HDIM;
    const float* __restrict__ w2 = W + (size_t)(colBase + 32 + l16) * HDIM;
    const float* __restrict__ w3 = W + (size_t)(colBase + 48 + l16) * HDIM;

    v8f acc[4] = {v8f{}, v8f{}, v8f{}, v8f{}};

#pragma unroll 4
    for (int k = 0; k < HDIM; k += 4) {
        const int kb = k + 2 * hi;
        // A fragment from LDS: lane l -> row = row0 + l%16, K = kb, kb+1
        v2f a = *(const v2f*)(arowp + kb);
        a *= scale;
        // B fragments: lane l -> col = colBase + 16*j + l%16, K = kb, kb+1
        v2f b0 = *(const v2f*)(w0 + kb);
        v2f b1 = *(const v2f*)(w1 + kb);
        v2f b2 = *(const v2f*)(w2 + kb);
        v2f b3 = *(const v2f*)(w3 + kb);
        acc[0] = __builtin_amdgcn_wmma_f32_16x16x4_f32(false, a, false, b0,
                                                       (short)0, acc[0], false, false);
        acc[1] = __builtin_amdgcn_wmma_f32_16x16x4_f32(false, a, false, b1,
                                                       (short)0, acc[1], false, false);
        acc[2] = __builtin_amdgcn_wmma_f32_16x16x4_f32(false, a, false, b2,
                                                       (short)0, acc[2], false, false);
        acc[3] = __builtin_amdgcn_wmma_f32_16x16x4_f32(false, a, false, b3,
                                                       (short)0, acc[3], false, false);
    }

    // C/D layout: vgpr v, lane l -> row = row0 + v + 8*(l/16), col = l%16
#pragma unroll
    for (int j = 0; j < 4; ++j) {
        const int col = colBase + 16 * j + l16;
        const float b = bias ? bias[col] : 0.0f;
#pragma unroll
        for (int v = 0; v < 8; ++v) {
            size_t idx = (size_t)(row0 + v + 8 * hi) * (size_t)Hout + (size_t)col;
            float val = acc[j][v] + b;
            if (accumulate) val += C[idx];
            C[idx] = val;
        }
    }
}

// ---------------- cosine similarity: one wave per row ----------------
__global__ void hg_cosine(const float* __restrict__ ro,
                          const float* __restrict__ no,
                          float* __restrict__ out, int n)
{
    int gw   = (int)((blockIdx.x * blockDim.x + threadIdx.x) >> 5);
    int lane = threadIdx.x & 31;
    if (gw >= n) return;
    const float* a = ro + (size_t)gw * OUTC;
    const float* b = no + (size_t)gw * OUTC;
    float dot = 0.f, na = 0.f, nb = 0.f;
#pragma unroll
    for (int c = lane; c < OUTC; c += 32) {
        float x = a[c], y = b[c];
        dot += x * y; na += x * x; nb += y * y;
    }
#pragma unroll
    for (int off = 16; off; off >>= 1) {
        dot += __shfl_xor(dot, off, 32);
        na  += __shfl_xor(na,  off, 32);
        nb  += __shfl_xor(nb,  off, 32);
    }
    if (lane == 0) {
        float d = fmaxf(sqrtf(na), 1e-8f) * fmaxf(sqrtf(nb), 1e-8f);
        out[gw] = (dot / d + 1.0f) * 0.5f;
    }
}

extern "C" void kernel_launch(void* const* d_in, const int* in_sizes, int n_in,
                              void* d_out, int out_size, void* d_ws, size_t ws_size,
                              hipStream_t stream) {
    const float* x_r   = (const float*)d_in[0];
    const float* x_p   = (const float*)d_in[1];
    const float* x_m   = (const float*)d_in[2];
    const float* notes = (const float*)d_in[3];
    const int* ei_p2r  = (const int*)d_in[4];
    const int* ei_m2r  = (const int*)d_in[5];
    const int* ei_r2p  = (const int*)d_in[6];
    const int* ei_r2m  = (const int*)d_in[7];
    const float* W_l   = (const float*)d_in[8];   // [L][R][H][H]
    const float* b_l   = (const float*)d_in[9];   // [L][R][H]
    const float* W_r   = (const float*)d_in[10];  // [L][R][H][H]
    const float* W_re  = (const float*)d_in[11];  // [OUT][H]
    const float* b_re  = (const float*)d_in[12];
    const float* W_no  = (const float*)d_in[13];
    const float* b_no  = (const float*)d_in[14];

    const int N = in_sizes[0] / HDIM;     // 20000
    const int E = in_sizes[4] / 2;        // 640000
    const size_t NH = (size_t)N * HDIM;

    float* ws = (float*)d_ws;
    float* bufA[3] = { ws,          ws + NH,     ws + 2 * NH };
    float* bufB[3] = { ws + 3 * NH, ws + 4 * NH, ws + 5 * NH };
    float* agg = ws + 6 * NH;
    float* deg = ws + 7 * NH;             // N floats
    float* ro  = deg + N;                 // N*OUTC
    float* no  = ro + (size_t)N * OUTC;   // N*OUTC

    const dim3 gGrid(N / 16);             // 1250 row-tiles
    const int scatterBlocks = (E + 7) / 8;  // 8 waves/block, 1 edge/wave

    #define WL(l, r) (W_l + (((size_t)(l) * RELS + (r)) * HDIM * HDIM))
    #define BL(l, r) (b_l + (((size_t)(l) * RELS + (r)) * HDIM))
    #define WR(l, r) (W_r + (((size_t)(l) * RELS + (r)) * HDIM * HDIM))

    auto neighbor_half = [&](const float* xsrc, const int* ei,
                             const float* Wl, const float* bl,
                             float* outp, int acc) {
        hipMemsetAsync(agg, 0, NH * sizeof(float), stream);
        hipMemsetAsync(deg, 0, (size_t)N * sizeof(float), stream);
        hg_scatter<<<scatterBlocks, 256, 0, stream>>>(xsrc, ei, ei + E, agg, deg, E);
        hg_invdeg<<<(N + 255) / 256, 256, 0, stream>>>(deg, N);
        hg_gemm_wmma<<<gGrid, 128, 0, stream>>>(agg, Wl, bl, deg, outp, HDIM, acc);
    };

    const float* cur[3] = { x_r, x_p, x_m };
    for (int l = 0; l < LAYERS; ++l) {
        float** nxt = (l & 1) ? bufB : bufA;

        // reaction <- protein (rel 0), then <- molecule (rel 1), HeteroConv sum
        neighbor_half(cur[1], ei_p2r, WL(l, 0), BL(l, 0), nxt[0], /*acc=*/0);
        hg_gemm_wmma<<<gGrid, 128, 0, stream>>>(cur[0], WR(l, 0), nullptr, nullptr,
                                                nxt[0], HDIM, /*acc=*/1);
        neighbor_half(cur[2], ei_m2r, WL(l, 1), BL(l, 1), nxt[0], /*acc=*/1);
        hg_gemm_wmma<<<gGrid, 128, 0, stream>>>(cur[0], WR(l, 1), nullptr, nullptr,
                                                nxt[0], HDIM, /*acc=*/1);

        // protein <- reaction (rel 2)
        neighbor_half(cur[0], ei_r2p, WL(l, 2), BL(l, 2), nxt[1], /*acc=*/0);
        hg_gemm_wmma<<<gGrid, 128, 0, stream>>>(cur[1], WR(l, 2), nullptr, nullptr,
                                                nxt[1], HDIM, /*acc=*/1);

        // molecule <- reaction (rel 3)
        neighbor_half(cur[0], ei_r2m, WL(l, 3), BL(l, 3), nxt[2], /*acc=*/0);
        hg_gemm_wmma<<<gGrid, 128, 0, stream>>>(cur[2], WR(l, 3), nullptr, nullptr,
                                                nxt[2], HDIM, /*acc=*/1);

        cur[0] = nxt[0]; cur[1] = nxt[1]; cur[2] = nxt[2];
    }

    // output heads: Hout = 128 -> 2 waves per block (64 threads)
    hg_gemm_wmma<<<gGrid, 64, 0, stream>>>(cur[0], W_re, b_re, nullptr, ro, OUTC, 0);
    hg_gemm_wmma<<<gGrid, 64, 0, stream>>>(notes,  W_no, b_no, nullptr, no, OUTC, 0);

    // cosine similarity, (cos+1)/2
    hg_cosine<<<(N + 7) / 8, 256, 0, stream>>>(ro, no, (float*)d_out, N);

    #undef WL
    #undef BL
    #undef WR
}